// SimpleHeteroGCN_34282428956831
// MI455X (gfx1250) — compile-verified
//
#include <hip/hip_runtime.h>

#define NN      100000      // nodes per type (user == item == 100000)
#define NEDGE   1600000
#define DIN     128
#define HH      64

typedef __attribute__((ext_vector_type(2))) float v2f;
typedef __attribute__((ext_vector_type(8))) float v8f;

// ---------------------------------------------------------------- degree prep
__global__ void k_fill1(float* __restrict__ p, int n) {
    int i = blockIdx.x * blockDim.x + threadIdx.x;
    if (i < n) p[i] = 1.0f;              // self-loop contributes 1 to degree
}

__global__ void k_deg(const int* __restrict__ dst, float* __restrict__ deg, int e) {
    int i = blockIdx.x * blockDim.x + threadIdx.x;
    if (i < e) unsafeAtomicAdd(&deg[dst[i]], 1.0f);
}

__global__ void k_rsqrt(float* __restrict__ p, int n) {
    int i = blockIdx.x * blockDim.x + threadIdx.x;
    if (i < n) p[i] = rsqrtf(p[i]);      // deg >= 1 always (self loop)
}

// ---------------------------------------------------------------- WMMA GEMM
// C[M,64] = A[M,K] @ W[K,64], f32 exact via V_WMMA_F32_16X16X4_F32.
// One wave -> 16 rows x 64 cols (4 N-tiles). M % 16 == 0 (100000 = 6250*16).
__global__ void k_gemm(const float* __restrict__ A, const float* __restrict__ W,
                       float* __restrict__ C, int M, int K) {
    int wave = blockIdx.x * (blockDim.x >> 5) + (threadIdx.x >> 5);
    int lane = threadIdx.x & 31;
    int m0 = wave << 4;
    if (m0 >= M) return;                 // wave-uniform: EXEC stays all-ones
    int g = lane >> 4;                   // lane group 0/1 -> K pair {0,1}/{2,3}
    int j = lane & 15;                   // row (A) / column (B,C) within tile

    v8f acc0 = {}, acc1 = {}, acc2 = {}, acc3 = {};
    const float* __restrict__ arow = A + (size_t)(m0 + j) * K;

    for (int k = 0; k < K; k += 4) {
        // A fragment 16x4: lane(g,j) holds A[m0+j][k+2g], A[m0+j][k+2g+1]
        v2f a;
        a.x = arow[k + 2 * g];
        a.y = arow[k + 2 * g + 1];
        // B fragments 4x16: lane(g,j) holds W[k+2g][n0+j], W[k+2g+1][n0+j]
        const float* __restrict__ w0 = W + (size_t)(k + 2 * g) * HH;
        const float* __restrict__ w1 = w0 + HH;
        v2f b0, b1, b2, b3;
        b0.x = w0[j];      b0.y = w1[j];
        b1.x = w0[16 + j]; b1.y = w1[16 + j];
        b2.x = w0[32 + j]; b2.y = w1[32 + j];
        b3.x = w0[48 + j]; b3.y = w1[48 + j];
        acc0 = __builtin_amdgcn_wmma_f32_16x16x4_f32(false, a, false, b0, (short)0, acc0, false, false);
        acc1 = __builtin_amdgcn_wmma_f32_16x16x4_f32(false, a, false, b1, (short)0, acc1, false, false);
        acc2 = __builtin_amdgcn_wmma_f32_16x16x4_f32(false, a, false, b2, (short)0, acc2, false, false);
        acc3 = __builtin_amdgcn_wmma_f32_16x16x4_f32(false, a, false, b3, (short)0, acc3, false, false);
    }
    // C/D layout: VGPR r, lanes 0-15 -> M=r, lanes 16-31 -> M=8+r, N=j
    int mbase = m0 + 8 * g;
    for (int r = 0; r < 8; ++r) {
        size_t off = (size_t)(mbase + r) * HH + j;
        C[off]      = acc0[r];
        C[off + 16] = acc1[r];
        C[off + 32] = acc2[r];
        C[off + 48] = acc3[r];
    }
}

// ---------------------------------------------------------------- aggregation
// out[i] = h[i] * dinv[i]^2   (the added self-loop term, also zero-initializes out)
__global__ void k_selfloop(const float* __restrict__ h, const float* __restrict__ dinv,
                           float* __restrict__ out, int n) {
    int idx = blockIdx.x * blockDim.x + threadIdx.x;
    if (idx >= n * HH) return;
    float di = dinv[idx >> 6];
    out[idx] = h[idx] * di * di;
}

// 64 threads per edge: coalesced gather of h[src], clustered atomics to out[dst]
__global__ void k_scatter(const int* __restrict__ src, const int* __restrict__ dst,
                          const float* __restrict__ dinv, const float* __restrict__ h,
                          float* __restrict__ out, int e) {
    long long idx = (long long)blockIdx.x * blockDim.x + threadIdx.x;
    if (idx >= (long long)e * HH) return;
    int ed = (int)(idx >> 6);
    int j  = (int)(idx & (HH - 1));
    int s = src[ed], d = dst[ed];
    float norm = dinv[s] * dinv[d];
    unsafeAtomicAdd(&out[(size_t)d * HH + j], h[(size_t)s * HH + j] * norm);
}

__global__ void k_bias_relu(float* __restrict__ p, const float* __restrict__ b, int n) {
    int idx = blockIdx.x * blockDim.x + threadIdx.x;
    if (idx >= n * HH) return;
    float v = p[idx] + b[idx & (HH - 1)];
    p[idx] = v > 0.0f ? v : 0.0f;
}

// ---------------------------------------------------------------- output head
__global__ void k_head(const float* __restrict__ h, const float* __restrict__ Wl,
                       const float* __restrict__ bl, float* __restrict__ out, int n) {
    int i = blockIdx.x * blockDim.x + threadIdx.x;
    if (i >= n) return;
    const float* __restrict__ row = h + (size_t)i * HH;
    float a0 = bl[0], a1 = bl[1];
    #pragma unroll
    for (int k = 0; k < HH; ++k) {
        float v = row[k];
        a0 += v * Wl[2 * k];
        a1 += v * Wl[2 * k + 1];
    }
    out[(size_t)i * 2]     = a0;
    out[(size_t)i * 2 + 1] = a1;
}

// ---------------------------------------------------------------- host driver
static void run_type(const float* x, const int* ei,
                     const float* W1, const float* b1,
                     const float* W2, const float* b2,
                     const float* Wl, const float* bl,
                     float* out, float* dinv, float* bufA, float* bufB,
                     hipStream_t stream) {
    const int T = 256;
    const int n = NN;
    int nodeBlk = (n + T - 1) / T;
    int featBlk = (n * HH + T - 1) / T;
    int edgeBlk = (NEDGE + T - 1) / T;
    long long tot = (long long)NEDGE * HH;
    int scatBlk = (int)((tot + T - 1) / T);
    int gemmBlk = (n / 16 + 7) / 8;      // 8 waves (of 32) per 256-thread block

    // degree -> dinv
    k_fill1 <<<nodeBlk, T, 0, stream>>>(dinv, n);
    k_deg   <<<edgeBlk, T, 0, stream>>>(ei + NEDGE, dinv, NEDGE);
    k_rsqrt <<<nodeBlk, T, 0, stream>>>(dinv, n);

    // layer 1: h = relu(agg(x @ W1) + b1)
    k_gemm     <<<gemmBlk, T, 0, stream>>>(x, W1, bufA, n, DIN);
    k_selfloop <<<featBlk, T, 0, stream>>>(bufA, dinv, bufB, n);
    k_scatter  <<<scatBlk, T, 0, stream>>>(ei, ei + NEDGE, dinv, bufA, bufB, NEDGE);
    k_bias_relu<<<featBlk, T, 0, stream>>>(bufB, b1, n);

    // layer 2: h = relu(agg(h @ W2) + b2)
    k_gemm     <<<gemmBlk, T, 0, stream>>>(bufB, W2, bufA, n, HH);
    k_selfloop <<<featBlk, T, 0, stream>>>(bufA, dinv, bufB, n);
    k_scatter  <<<scatBlk, T, 0, stream>>>(ei, ei + NEDGE, dinv, bufA, bufB, NEDGE);
    k_bias_relu<<<featBlk, T, 0, stream>>>(bufB, b2, n);

    // head: out = h @ Wl + bl
    k_head <<<nodeBlk, T, 0, stream>>>(bufB, Wl, bl, out, n);
}

extern "C" void kernel_launch(void* const* d_in, const int* in_sizes, int n_in,
                              void* d_out, int out_size, void* d_ws, size_t ws_size,
                              hipStream_t stream) {
    const float* x_user  = (const float*)d_in[0];
    const float* x_item  = (const float*)d_in[1];
    const int*   ei_user = (const int*)d_in[2];   // [0..E): src, [E..2E): dst
    const int*   ei_item = (const int*)d_in[3];
    const float* W1_u = (const float*)d_in[4];
    const float* b1_u = (const float*)d_in[5];
    const float* W1_i = (const float*)d_in[6];
    const float* b1_i = (const float*)d_in[7];
    const float* W2_u = (const float*)d_in[8];
    const float* b2_u = (const float*)d_in[9];
    const float* W2_i = (const float*)d_in[10];
    const float* b2_i = (const float*)d_in[11];
    const float* Wl_u = (const float*)d_in[12];
    const float* bl_u = (const float*)d_in[13];
    const float* Wl_i = (const float*)d_in[14];
    const float* bl_i = (const float*)d_in[15];

    float* out = (float*)d_out;          // [user 100000x2][item 100000x2]

    // workspace: dinv (100k, padded), bufA (100k x 64), bufB (100k x 64) ~52 MB
    float* dinv = (float*)d_ws;
    float* bufA = dinv + ((NN + 255) & ~255);
    float* bufB = bufA + (size_t)NN * HH;

    run_type(x_user, ei_user, W1_u, b1_u, W2_u, b2_u, Wl_u, bl_u,
             out, dinv, bufA, bufB, stream);
    run_type(x_item, ei_item, W1_i, b1_i, W2_i, b2_i, Wl_i, bl_i,
             out + (size_t)NN * 2, dinv, bufA, bufB, stream);
}